// OptNet_26104811225850
// MI455X (gfx1250) — compile-verified
//
#include <hip/hip_runtime.h>
#include <math.h>

// ---------------------------------------------------------------------------
// OptNet forward for MI455X (gfx1250, wave32, WMMA, async global->LDS).
//   x(256,2048) -> GEMM1+relu -> BN -> GEMM2+relu -> BN -> p(256,128)
//   Q = tril(L)tril(L)^T + 1e-4 I ;  h = G z0 + s0
//   15 Mehrotra predictor-corrector IPM iterations (one workgroup per batch;
//   G staged to LDS via GLOBAL_LOAD_ASYNC_TO_LDS_B128; H/chol in LDS;
//   triangular solves register-resident in one wave32), then log_softmax.
// MLP GEMM: ASYNCcnt-double-buffered LDS tiles feeding WMMA from ds_load.
// All matrix products use V_WMMA_F32_16X16X4_F32 (f32 kept for IPM stability).
// ---------------------------------------------------------------------------

typedef float v2f __attribute__((ext_vector_type(2)));
typedef float v8f __attribute__((ext_vector_type(8)));

#define WMMA_F32(a, b, c) \
  __builtin_amdgcn_wmma_f32_16x16x4_f32(false, (a), false, (b), (short)0, (c), false, false)

constexpr int   BX      = 256;    // batch
constexpr int   NF      = 2048;   // features
constexpr int   NH      = 2048;   // hidden
constexpr int   NC      = 128;    // classes (QP dim n)
constexpr int   NI      = 200;    // inequalities (QP dim m)
constexpr int   LDH     = 129;    // padded LDS leading dim for H
constexpr int   NITERS  = 15;
constexpr float BN_EPS_C = 1e-5f;
constexpr float QP_EPS_C = 1e-4f;

// GEMM block tiling
constexpr int   KB = 32;          // k-block depth (floats)
constexpr int   MB = 64;          // block rows   (one 64-row stripe)
constexpr int   NB = 128;         // block cols   (8 waves x 16)

// ---------------------------------------------------------------------------
// C(M,N) = relu(A(M,K) @ W(N,K)^T + bias).
// Block = 8 wave32 computing a 64x128 tile; wave w owns the 64x16 slab at
// column w*16.  Per 32-deep k-block, A(64x32) and W(128x32) are staged into
// LDS with GLOBAL_LOAD_ASYNC_TO_LDS_B128 (6 async ops per wave per buffer),
// double-buffered on ASYNCcnt: issue k+1, s_wait_asynccnt 6 (old buffer
// complete, new 6 still in flight), barrier, 32 WMMAs from ds_load_b64.
// A-frag (16x4 f32): lanes 0-15 hold K = k,k+1 ; lanes 16-31 hold K = k+2,k+3.
// ---------------------------------------------------------------------------
__global__ __launch_bounds__(256) void gemm_bias_relu_f32(
    const float* __restrict__ A, const float* __restrict__ W,
    const float* __restrict__ bias, float* __restrict__ C,
    int M, int N, int K)
{
  __shared__ float sA[2][MB * KB];                     // 2 x 8 KB
  __shared__ float sW[2][NB * KB];                     // 2 x 16 KB
  const int t    = threadIdx.x;
  const int wave = t >> 5, lane = t & 31, half = lane >> 4, l16 = lane & 15;
  const int nbn  = N / NB;
  const int bm   = (int)blockIdx.x / nbn;
  const int bn   = (int)blockIdx.x % nbn;
  const int m0   = bm * MB;
  const int n0b  = bn * NB;
  (void)M;

  // stage one k-block into LDS buffer `buf` (A: 512, W: 1024 16B chunks)
  auto stage = [&](int buf, int k0) {
    for (int c = t; c < (MB * KB) / 4; c += 256) {     // 2 per thread
      const int row = c >> 3, cq = c & 7;
      const unsigned goff = (unsigned)(((m0 + row) * K + k0 + cq * 4) * 4);
      const unsigned lds  = (unsigned)(size_t)&sA[buf][row * KB + cq * 4];
      asm volatile("global_load_async_to_lds_b128 %0, %1, %2 offset:0"
                   :: "v"(lds), "v"(goff), "s"(A) : "memory");
    }
    for (int c = t; c < (NB * KB) / 4; c += 256) {     // 4 per thread
      const int row = c >> 3, cq = c & 7;
      const unsigned goff = (unsigned)(((n0b + row) * K + k0 + cq * 4) * 4);
      const unsigned lds  = (unsigned)(size_t)&sW[buf][row * KB + cq * 4];
      asm volatile("global_load_async_to_lds_b128 %0, %1, %2 offset:0"
                   :: "v"(lds), "v"(goff), "s"(W) : "memory");
    }
  };

  v8f acc0 = {}, acc1 = {}, acc2 = {}, acc3 = {};
  const int nkb = K / KB;
  stage(0, 0);
  for (int kb = 0; kb < nkb; ++kb) {
    if (kb + 1 < nkb) {
      stage((kb + 1) & 1, (kb + 1) * KB);              // prefetch next buffer
      asm volatile("s_wait_asynccnt 6" ::: "memory");  // current buffer landed
    } else {
      asm volatile("s_wait_asynccnt 0" ::: "memory");
    }
    __syncthreads();

    const float* pa = sA[kb & 1];
    const float* pw = sW[kb & 1] + (wave * 16 + l16) * KB;
#pragma unroll
    for (int ks = 0; ks < KB; ks += 4) {
      const int kk = ks + 2 * half;
      v2f b, a;
      b.x = pw[kk];                   b.y = pw[kk + 1];
      a.x = pa[(l16     ) * KB + kk]; a.y = pa[(l16     ) * KB + kk + 1];
      acc0 = WMMA_F32(a, b, acc0);
      a.x = pa[(l16 + 16) * KB + kk]; a.y = pa[(l16 + 16) * KB + kk + 1];
      acc1 = WMMA_F32(a, b, acc1);
      a.x = pa[(l16 + 32) * KB + kk]; a.y = pa[(l16 + 32) * KB + kk + 1];
      acc2 = WMMA_F32(a, b, acc2);
      a.x = pa[(l16 + 48) * KB + kk]; a.y = pa[(l16 + 48) * KB + kk + 1];
      acc3 = WMMA_F32(a, b, acc3);
    }
    __syncthreads();                                   // buffer reuse fence
  }

  const int n0 = n0b + wave * 16;
  const float bnv = bias[n0 + l16];
#pragma unroll
  for (int r = 0; r < 8; ++r) {
    const int row = m0 + r + 8 * half;
    C[(size_t)(row     ) * N + n0 + l16] = fmaxf(acc0[r] + bnv, 0.f);
    C[(size_t)(row + 16) * N + n0 + l16] = fmaxf(acc1[r] + bnv, 0.f);
    C[(size_t)(row + 32) * N + n0 + l16] = fmaxf(acc2[r] + bnv, 0.f);
    C[(size_t)(row + 48) * N + n0 + l16] = fmaxf(acc3[r] + bnv, 0.f);
  }
}

// ---------------------------------------------------------------------------
// Training-mode BatchNorm1d over columns; one block (256 thr) per column.
// ---------------------------------------------------------------------------
__global__ __launch_bounds__(256) void batchnorm_col(
    float* __restrict__ X, const float* __restrict__ gamma,
    const float* __restrict__ beta, int M, int N)
{
  __shared__ float red[256];
  const int col = blockIdx.x;
  const int t   = threadIdx.x;
  float v = (t < M) ? X[(size_t)t * N + col] : 0.f;
  red[t] = v; __syncthreads();
  for (int s = 128; s > 0; s >>= 1) { if (t < s) red[t] += red[t + s]; __syncthreads(); }
  const float mu = red[0] / (float)M; __syncthreads();
  const float d = (t < M) ? (v - mu) : 0.f;
  red[t] = d * d; __syncthreads();
  for (int s = 128; s > 0; s >>= 1) { if (t < s) red[t] += red[t + s]; __syncthreads(); }
  const float var = red[0] / (float)M;
  if (t < M)
    X[(size_t)t * N + col] = gamma[col] * d * rsqrtf(var + BN_EPS_C) + beta[col];
}

// ---------------------------------------------------------------------------
// Q = tril(L) @ tril(L)^T + eps*I   (128x128, K=128).  64 tiles, grid 8.
// ---------------------------------------------------------------------------
__global__ __launch_bounds__(256) void build_Q(const float* __restrict__ L,
                                               float* __restrict__ Q)
{
  const int wave = threadIdx.x >> 5, lane = threadIdx.x & 31;
  const int half = lane >> 4, l16 = lane & 15;
  const int tile = blockIdx.x * 8 + wave;
  const int i0 = (tile >> 3) << 4, j0 = (tile & 7) << 4;
  const int row = i0 + l16, col = j0 + l16;
  v8f acc = {};
  for (int k = 0; k < NC; k += 4) {
    const int ka = k + 2 * half;
    v2f a, b;
    a.x = (ka     <= row) ? L[row * NC + ka]     : 0.f;
    a.y = (ka + 1 <= row) ? L[row * NC + ka + 1] : 0.f;
    b.x = (ka     <= col) ? L[col * NC + ka]     : 0.f;   // tril(L)^T(k,j)=L[j,k]
    b.y = (ka + 1 <= col) ? L[col * NC + ka + 1] : 0.f;
    acc = WMMA_F32(a, b, acc);
  }
#pragma unroll
  for (int r = 0; r < 8; ++r) {
    const int i = i0 + r + 8 * half;
    Q[i * NC + j0 + l16] = acc[r] + ((i == j0 + l16) ? QP_EPS_C : 0.f);
  }
}

// ------------------------- block-wide reductions ---------------------------
__device__ __forceinline__ float blk_sum(float v, float* red, int t) {
  red[t] = v; __syncthreads();
  for (int s = 128; s > 0; s >>= 1) { if (t < s) red[t] += red[t + s]; __syncthreads(); }
  const float r = red[0]; __syncthreads(); return r;
}
__device__ __forceinline__ float blk_min(float v, float* red, int t) {
  red[t] = v; __syncthreads();
  for (int s = 128; s > 0; s >>= 1) { if (t < s) red[t] = fminf(red[t], red[t + s]); __syncthreads(); }
  const float r = red[0]; __syncthreads(); return r;
}

// largest a with v + a*dv >= 0 (min over m)
__device__ __forceinline__ float max_step_dev(const float* v, const float* dv,
                                              float* red, int t) {
  float loc = INFINITY;
  for (int m = t; m < NI; m += 256)
    if (dv[m] < 0.f) loc = fminf(loc, -v[m] / dv[m]);
  return blk_min(loc, red, t);
}

// ---------------------------------------------------------------------------
// chol(H) \ rhs (fwd) then chol(H)^T \ y (bwd) — vector held in registers of
// wave 0 (4 elems/lane), broadcast via __shfl; NO block barriers inside.
// ---------------------------------------------------------------------------
__device__ __forceinline__ void cho_solve_wave0(const float* sH, float* dz,
                                                int t)
{
  const int wave = t >> 5, lane = t & 31;
  if (wave == 0) {
    float y0 = dz[lane], y1 = dz[lane + 32], y2 = dz[lane + 64], y3 = dz[lane + 96];
    const int i1 = lane + 32, i2 = lane + 64, i3 = lane + 96;
    // forward:  L y = rhs
    for (int k = 0; k < NC; ++k) {
      const int kr = k >> 5, kl = k & 31;
      float part = (kr == 0) ? y0 : (kr == 1) ? y1 : (kr == 2) ? y2 : y3;
      const float yk = __shfl(part, kl, 32) / sH[k * LDH + k];
      if (lane > k) y0 -= sH[lane * LDH + k] * yk;
      if (i1   > k) y1 -= sH[i1   * LDH + k] * yk;
      if (i2   > k) y2 -= sH[i2   * LDH + k] * yk;
      if (i3   > k) y3 -= sH[i3   * LDH + k] * yk;
      if (lane == kl) {
        if (kr == 0) y0 = yk; else if (kr == 1) y1 = yk;
        else if (kr == 2) y2 = yk; else y3 = yk;
      }
    }
    // backward: L^T x = y
    for (int k = NC - 1; k >= 0; --k) {
      const int kr = k >> 5, kl = k & 31;
      float part = (kr == 0) ? y0 : (kr == 1) ? y1 : (kr == 2) ? y2 : y3;
      const float xk = __shfl(part, kl, 32) / sH[k * LDH + k];
      if (lane < k) y0 -= sH[k * LDH + lane] * xk;
      if (i1   < k) y1 -= sH[k * LDH + i1  ] * xk;
      if (i2   < k) y2 -= sH[k * LDH + i2  ] * xk;
      if (i3   < k) y3 -= sH[k * LDH + i3  ] * xk;
      if (lane == kl) {
        if (kr == 0) y0 = xk; else if (kr == 1) y1 = xk;
        else if (kr == 2) y2 = xk; else y3 = xk;
      }
    }
    dz[lane] = y0; dz[lane + 32] = y1; dz[lane + 64] = y2; dz[lane + 96] = y3;
  }
  __syncthreads();
}

// ------------------------- one Newton solve (per batch) --------------------
__device__ void newton_dev(const float* sG,
                           const float* sH, const float* rd, const float* rp,
                           const float* sv, const float* lamv,
                           const float* cent, float* uv,
                           float* dz, float* dsv, float* dlv, int t)
{
  for (int m = t; m < NI; m += 256)
    uv[m] = (lamv[m] * rp[m] - cent[m]) / sv[m];
  __syncthreads();
  if (t < NC) {
    float acc = -rd[t];
    for (int m = 0; m < NI; ++m) acc -= uv[m] * sG[m * NC + t];
    dz[t] = acc;
  }
  __syncthreads();
  cho_solve_wave0(sH, dz, t);
  for (int m = t; m < NI; m += 256) {
    float acc = -rp[m];
    const float* Gr = sG + m * NC;
    for (int i = 0; i < NC; ++i) acc -= dz[i] * Gr[i];
    dsv[m] = acc;
    dlv[m] = (-cent[m] - lamv[m] * acc) / sv[m];
  }
  __syncthreads();
}

// ---------------------------------------------------------------------------
// Full 15-iteration IPM: one workgroup (8 wave32) per batch element.
// G (100 KB) async-staged into LDS once; Q streams from L2 (prefetched);
// H/chol + all solver vectors live in LDS (~176 KB of the 320 KB WGP budget).
// ---------------------------------------------------------------------------
__global__ __launch_bounds__(256) void qp_solve_kernel(
    const float* __restrict__ Qg, const float* __restrict__ Pg,
    const float* __restrict__ Gg, const float* __restrict__ z0g,
    const float* __restrict__ s0g, float* __restrict__ out)
{
  __shared__ float sG[NI * NC];                        // staged G (row-major)
  __shared__ float sH[NC * LDH];                       // H then chol in place
  __shared__ float zv[NC], rd[NC], dz[NC];
  __shared__ float sv[NI], lamv[NI], rp[NI], wv[NI], cent[NI], uv[NI];
  __shared__ float dsv[NI], dlv[NI], dsa[NI], dla[NI], hv[NI];
  __shared__ float red[256];

  const int t    = threadIdx.x;
  const int b    = blockIdx.x;
  const int wave = t >> 5, lane = t & 31, half = lane >> 4, l16 = lane & 15;

  // ----- async stage G into LDS: 6400 x 16B chunks, ASYNCcnt-tracked -----
  for (int idx = t; idx < (NI * NC) / 4; idx += 256) {
    const unsigned lds_addr = (unsigned)(size_t)(&sG[idx * 4]);
    const unsigned goff     = (unsigned)(idx * 16);
    asm volatile("global_load_async_to_lds_b128 %0, %1, %2 offset:0"
                 :: "v"(lds_addr), "v"(goff), "s"(Gg) : "memory");
  }
  asm volatile("s_wait_asynccnt 0" ::: "memory");
  __syncthreads();

  // ----- init: z=0, s=1, lam=1, h = G z0 + s0 -----
  if (t < NC) zv[t] = 0.f;
  for (int m = t; m < NI; m += 256) {
    sv[m] = 1.f; lamv[m] = 1.f;
    float acc = s0g[m];
    const float* Gr = sG + m * NC;
    for (int i = 0; i < NC; ++i) acc += Gr[i] * z0g[i];
    hv[m] = acc;
  }
  __syncthreads();

  for (int it = 0; it < NITERS; ++it) {
    // ----- residuals -----
    if (t < NC) {
      __builtin_prefetch(&Qg[t * NC], 0, 1);           // global_prefetch_b8
      float acc = Pg[b * NC + t];
      for (int i = 0; i < NC; ++i) acc += zv[i] * Qg[i * NC + t];
      for (int m = 0; m < NI; ++m) acc += lamv[m] * sG[m * NC + t];
      rd[t] = acc;
    }
    for (int m = t; m < NI; m += 256) {
      float acc = sv[m] - hv[m];
      const float* Gr = sG + m * NC;
      for (int i = 0; i < NC; ++i) acc += zv[i] * Gr[i];
      rp[m] = acc;
    }
    // ----- mu and w -----
    float part = 0.f;
    for (int m = t; m < NI; m += 256) part += lamv[m] * sv[m];
    const float mu = fmaxf(blk_sum(part, red, t) / (float)NI, 1e-12f);
    for (int m = t; m < NI; m += 256) wv[m] = lamv[m] / sv[m];
    __syncthreads();

    // ----- H = Q + G^T diag(w) G  (WMMA over K=200, from LDS) -----
    for (int tile = wave; tile < 64; tile += 8) {
      const int i0 = (tile >> 3) << 4, j0 = (tile & 7) << 4;
      v8f acc = {};
      for (int m0 = 0; m0 < NI; m0 += 4) {
        const int ma = m0 + 2 * half;
        v2f a, bb;
        a.x  = sG[ma * NC + i0 + l16];                 // G^T(i, m) = G(m, i)
        a.y  = sG[(ma + 1) * NC + i0 + l16];
        bb.x = wv[ma]     * sG[ma * NC + j0 + l16];    // (diag(w)G)(m, j)
        bb.y = wv[ma + 1] * sG[(ma + 1) * NC + j0 + l16];
        acc = WMMA_F32(a, bb, acc);
      }
#pragma unroll
      for (int r = 0; r < 8; ++r) {
        const int i = i0 + r + 8 * half;
        sH[i * LDH + j0 + l16] = acc[r] + Qg[i * NC + j0 + l16];
      }
    }
    __syncthreads();

    // ----- in-place Cholesky (lower) -----
    for (int k = 0; k < NC; ++k) {
      if (t == 0) sH[k * LDH + k] = sqrtf(sH[k * LDH + k]);
      __syncthreads();
      const float dk = sH[k * LDH + k];
      for (int i = k + 1 + t; i < NC; i += 256) sH[i * LDH + k] /= dk;
      __syncthreads();
      for (int j = k + 1; j < NC; ++j) {
        const float ljk = sH[j * LDH + k];
        for (int i = j + t; i < NC; i += 256)
          sH[i * LDH + j] -= sH[i * LDH + k] * ljk;
      }
      __syncthreads();
    }

    // ----- predictor (affine) -----
    for (int m = t; m < NI; m += 256) cent[m] = lamv[m] * sv[m];
    __syncthreads();
    newton_dev(sG, sH, rd, rp, sv, lamv, cent, uv, dz, dsv, dlv, t);
    for (int m = t; m < NI; m += 256) { dsa[m] = dsv[m]; dla[m] = dlv[m]; }
    __syncthreads();
    const float a_aff = fminf(1.f, fminf(max_step_dev(lamv, dla, red, t),
                                         max_step_dev(sv,   dsa, red, t)));
    float pa = 0.f;
    for (int m = t; m < NI; m += 256)
      pa += (lamv[m] + a_aff * dla[m]) * (sv[m] + a_aff * dsa[m]);
    const float mu_aff = blk_sum(pa, red, t) / (float)NI;
    const float ratio  = mu_aff / mu;
    const float sigma  = ratio * ratio * ratio;

    // ----- corrector -----
    for (int m = t; m < NI; m += 256)
      cent[m] = lamv[m] * sv[m] + dla[m] * dsa[m] - sigma * mu;
    __syncthreads();
    newton_dev(sG, sH, rd, rp, sv, lamv, cent, uv, dz, dsv, dlv, t);
    const float alpha = fminf(1.f, 0.99f * fminf(max_step_dev(lamv, dlv, red, t),
                                                 max_step_dev(sv,   dsv, red, t)));
    if (t < NC) zv[t] += alpha * dz[t];
    for (int m = t; m < NI; m += 256) {
      sv[m]   += alpha * dsv[m];
      lamv[m] += alpha * dlv[m];
    }
    __syncthreads();
  }

  // ----- log_softmax over z -----
  red[t] = (t < NC) ? zv[t] : -INFINITY; __syncthreads();
  for (int s = 128; s > 0; s >>= 1) { if (t < s) red[t] = fmaxf(red[t], red[t + s]); __syncthreads(); }
  const float mx = red[0]; __syncthreads();
  const float ex = (t < NC) ? expf(zv[t] - mx) : 0.f;
  const float se = blk_sum(ex, red, t);
  if (t < NC) out[b * NC + t] = zv[t] - mx - logf(se);
}

// ---------------------------------------------------------------------------
extern "C" void kernel_launch(void* const* d_in, const int* in_sizes, int n_in,
                              void* d_out, int out_size, void* d_ws, size_t ws_size,
                              hipStream_t stream)
{
  (void)in_sizes; (void)n_in; (void)out_size; (void)ws_size;
  const float* x   = (const float*)d_in[0];
  const float* W1  = (const float*)d_in[1];
  const float* b1  = (const float*)d_in[2];
  const float* W2  = (const float*)d_in[3];
  const float* b2  = (const float*)d_in[4];
  const float* g1  = (const float*)d_in[5];
  const float* be1 = (const float*)d_in[6];
  const float* g2  = (const float*)d_in[7];
  const float* be2 = (const float*)d_in[8];
  const float* L   = (const float*)d_in[9];
  const float* G   = (const float*)d_in[10];
  const float* z0  = (const float*)d_in[11];
  const float* s0  = (const float*)d_in[12];
  float* out = (float*)d_out;

  float* h1 = (float*)d_ws;                      // 256*2048 f32 (2 MB)
  float* p  = h1 + (size_t)BX * NH;              // 256*128  f32
  float* Qm = p  + (size_t)BX * NC;              // 128*128  f32

  // MLP: 64x128 block tiles, async-LDS double-buffered WMMA GEMM
  gemm_bias_relu_f32<<<dim3((BX/MB)*(NH/NB)), dim3(256), 0, stream>>>(x,  W1, b1, h1, BX, NH, NF);
  batchnorm_col      <<<dim3(NH),             dim3(256), 0, stream>>>(h1, g1, be1, BX, NH);
  gemm_bias_relu_f32<<<dim3((BX/MB)*(NC/NB)), dim3(256), 0, stream>>>(h1, W2, b2, p,  BX, NC, NH);
  batchnorm_col      <<<dim3(NC),             dim3(256), 0, stream>>>(p,  g2, be2, BX, NC);

  // QP data + fused interior-point solve + log_softmax
  build_Q        <<<dim3(8),  dim3(256), 0, stream>>>(L, Qm);
  qp_solve_kernel<<<dim3(BX), dim3(256), 0, stream>>>(Qm, p, G, z0, s0, out);
}